// VAE_26482768347179
// MI455X (gfx1250) — compile-verified
//
#include <hip/hip_runtime.h>
#include <hip/hip_bf16.h>
#include <stdint.h>

typedef __attribute__((ext_vector_type(2))) float v2f;
typedef __attribute__((ext_vector_type(8))) float v8f;

#define HIDDEN 256
#define LATENT 128
#define VOCAB  32000
#define MAX_LEN 64
#define MAX_STEPS 128
#define N_NODES 511

// CDNA5 async global->LDS copy (ASYNCcnt-tracked), per ISA §10 / §15.18.3.
// lds_off is a byte offset into the workgroup's LDS; gptr a global address.
__device__ __forceinline__ void async_copy16_to_lds(unsigned lds_off, const void* gptr) {
    asm volatile("global_load_async_to_lds_b128 %0, %1, off"
                 :: "v"(lds_off), "v"(gptr) : "memory");
}
__device__ __forceinline__ void wait_async0() {
    asm volatile("s_wait_asynccnt 0" ::: "memory");
}

// ---------------------------------------------------------------------------
// Kernel 0: zero the output buffer (harness poisons it; inactive rows must be 0)
// ---------------------------------------------------------------------------
__global__ __launch_bounds__(256) void k_zero(float* __restrict__ p, int n) {
    int i = blockIdx.x * 256 + threadIdx.x;
    if (i < n) p[i] = 0.0f;
}

// ---------------------------------------------------------------------------
// Kernel 1: encoder, one tree level per launch.  One block per node; leaves
// copy the embedding, internal nodes do h[i]=Wpar@[emb,h_l,h_r]+bpar (256x768).
// ---------------------------------------------------------------------------
__global__ __launch_bounds__(256) void k_enc_level(
    const float* __restrict__ embed, const int* __restrict__ values,
    const int* __restrict__ left, const int* __restrict__ right,
    const float* __restrict__ Wpar, const float* __restrict__ bpar,
    float* __restrict__ h, int first)
{
    const int node = first + blockIdx.x;
    const int t = threadIdx.x;                  // 0..255 (output dim)
    const float* v = embed + (size_t)values[node] * HIDDEN;
    if (left[node] < 0) {                       // leaf (block-uniform branch)
        h[(size_t)node * HIDDEN + t] = v[t];
        return;
    }
    __shared__ __align__(16) float cat[3 * HIDDEN];
    const float* hl = h + (size_t)left[node]  * HIDDEN;
    const float* hr = h + (size_t)right[node] * HIDDEN;
    cat[t] = v[t]; cat[HIDDEN + t] = hl[t]; cat[2 * HIDDEN + t] = hr[t];
    __syncthreads();
    const float* wr = Wpar + (size_t)t * (3 * HIDDEN);
    float acc = bpar[t];
#pragma unroll 4
    for (int k = 0; k < 3 * HIDDEN; k += 4) {
        float4 w = *(const float4*)(wr + k);
        float4 c = *(const float4*)(cat + k);
        acc += w.x * c.x + w.y * c.y + w.z * c.z + w.w * c.w;
    }
    h[(size_t)node * HIDDEN + t] = acc;
}

// ---------------------------------------------------------------------------
// Kernel 2: latent head: mu, logvar, z = mu + eps*exp(0.5*logvar)
// ---------------------------------------------------------------------------
__global__ __launch_bounds__(256) void k_latent(
    const float* __restrict__ h,
    const float* __restrict__ Wmu, const float* __restrict__ bmu,
    const float* __restrict__ Wlv, const float* __restrict__ blv,
    const float* __restrict__ eps,
    float* __restrict__ mu_out, float* __restrict__ lv_out, float* __restrict__ z)
{
    __shared__ __align__(16) float root[HIDDEN];
    __shared__ float muS[LATENT], lvS[LATENT];
    const int t = threadIdx.x;
    root[t] = h[t];                              // root = h[0]
    __syncthreads();
    if (t < LATENT) {
        const float* wr = Wmu + (size_t)t * HIDDEN;
        float a = bmu[t];
#pragma unroll 4
        for (int k = 0; k < HIDDEN; k += 4) {
            float4 w = *(const float4*)(wr + k);
            float4 r = *(const float4*)(root + k);
            a += w.x * r.x + w.y * r.y + w.z * r.z + w.w * r.w;
        }
        muS[t] = a; mu_out[t] = a;
    } else {
        const int j = t - LATENT;
        const float* wr = Wlv + (size_t)j * HIDDEN;
        float a = blv[j];
#pragma unroll 4
        for (int k = 0; k < HIDDEN; k += 4) {
            float4 w = *(const float4*)(wr + k);
            float4 r = *(const float4*)(root + k);
            a += w.x * r.x + w.y * r.y + w.z * r.z + w.w * r.w;
        }
        lvS[j] = a; lv_out[j] = a;
    }
    __syncthreads();
    if (t < LATENT) z[t] = muS[t] + eps[t] * expf(0.5f * lvS[t]);
}

// ---------------------------------------------------------------------------
// Kernel 3: sequential decoder control chain (single workgroup stack machine).
// Records per-step co[1:257] into masked Cp (parent-active) / Cl (leaf-active)
// plus kind/op; computes the 256-row Wp tail for pushed latents.
// ---------------------------------------------------------------------------
__global__ __launch_bounds__(256) void k_dec_control(
    const float* __restrict__ W1, const float* __restrict__ b1,
    const float* __restrict__ W2, const float* __restrict__ b2,
    const float* __restrict__ WpTail, const float* __restrict__ bpTail, // Wp+32000*256
    const float* __restrict__ z,
    float* __restrict__ Cp, float* __restrict__ Cl,
    int* __restrict__ kind, int* __restrict__ opArr)
{
    __shared__ __align__(16) float stk[MAX_LEN][LATENT];   // 32 KB
    __shared__ __align__(16) float nodeb[LATENT];
    __shared__ __align__(16) float hmid[HIDDEN];
    __shared__ __align__(16) float co1[HIDDEN];            // co[1..256]
    __shared__ float co0;
    __shared__ int sp_s, op_s;
    const int t = threadIdx.x;

    if (t < LATENT) stk[0][t] = z[t];
    if (t == 0) { sp_s = 1; op_s = 0; }
    __syncthreads();

    for (int step = 0; step < MAX_STEPS; ++step) {
        const int sp = sp_s;
        const int op = op_s;
        const bool active = sp > 0;
        const int idx = active ? sp - 1 : 0;
        if (t < LATENT) nodeb[t] = stk[idx][t];
        __syncthreads();

        // hmid = leaky_relu(W1 @ node + b1, 0.2)
        {
            const float* wr = W1 + (size_t)t * LATENT;
            float a = b1[t];
#pragma unroll 4
            for (int k = 0; k < LATENT; k += 4) {
                float4 w = *(const float4*)(wr + k);
                float4 nv = *(const float4*)(nodeb + k);
                a += w.x * nv.x + w.y * nv.y + w.z * nv.z + w.w * nv.w;
            }
            hmid[t] = a > 0.0f ? a : 0.2f * a;
        }
        __syncthreads();

        // co = W2 @ hmid + b2   (257 outs; thread 0 also does out 256)
        for (int o = t; o < HIDDEN + 1; o += 256) {
            const float* wr = W2 + (size_t)o * HIDDEN;
            float a = b2[o];
#pragma unroll 4
            for (int k = 0; k < HIDDEN; k += 4) {
                float4 w = *(const float4*)(wr + k);
                float4 hv = *(const float4*)(hmid + k);
                a += w.x * hv.x + w.y * hv.y + w.z * hv.z + w.w * hv.w;
            }
            if (o == 0) co0 = a; else co1[o - 1] = a;
        }
        __syncthreads();

        const bool parent = (co0 > 0.0f) && (sp - 1 <= MAX_LEN - 2) && active;

        const float cv = co1[t];
        Cp[(size_t)step * HIDDEN + t] = (active && parent)  ? cv : 0.0f;
        Cl[(size_t)step * HIDDEN + t] = (active && !parent) ? cv : 0.0f;
        if (t == 0) {
            kind[step]  = active ? (parent ? 2 : 1) : 0;
            opArr[step] = op;
        }

        if (parent) {   // block-uniform: pushed latents ll/rl via Wp tail
            const float* wr = WpTail + (size_t)t * HIDDEN;
            float a = bpTail[t];
#pragma unroll 4
            for (int k = 0; k < HIDDEN; k += 4) {
                float4 w = *(const float4*)(wr + k);
                float4 cvv = *(const float4*)(co1 + k);
                a += w.x * cvv.x + w.y * cvv.y + w.z * cvv.z + w.w * cvv.w;
            }
            if (t < LATENT) stk[idx][t] = a;               // ll
            else            stk[idx + 1][t - LATENT] = a;  // rl
        }
        __syncthreads();
        if (t == 0 && active) {
            sp_s = parent ? sp + 1 : sp - 1;
            op_s = op + 1;
        }
        __syncthreads();
    }
}

// ---------------------------------------------------------------------------
// Kernel 4: vocab projection GEMM (M=128, N=32000, K=256) with f32 WMMA.
//   D[t][n] = (Cp[t].Wp[n]) + (Cl[t].Wl[n]);  out[op[t]][n] = D + bias(kind)
// A (Cp/Cl) is staged per-workgroup in LDS in 32-column K-chunks via
// global_load_async_to_lds_b128 (ASYNCcnt), cutting L2 A-traffic 8x; 8 waves
// per block each own one 16-column vocab tile and 8 step-tiles, reusing each
// B slice (one b64/lane from L2-resident weights) across 8 WMMAs.
// LDS row stride 36 floats: 16B-aligned rows + conflict-free ds_load_b64.
// ---------------------------------------------------------------------------
#define KCH 32          // K-chunk columns
#define ASTRIDE 36      // padded LDS row stride (floats)

__global__ __launch_bounds__(256) void k_vocab_gemm(
    const float* __restrict__ Wl, const float* __restrict__ bl,
    const float* __restrict__ Wp, const float* __restrict__ bp,
    const float* __restrict__ Cp, const float* __restrict__ Cl,
    const int* __restrict__ kind, const int* __restrict__ opArr,
    float* __restrict__ out)
{
    __shared__ __align__(16) float sA[2][MAX_STEPS * ASTRIDE];  // [Cp|Cl] chunk, ~36 KB

    const int tid   = threadIdx.x;
    const int lane  = tid & 31;
    const int wave  = tid >> 5;                    // 0..7
    const int ntile = blockIdx.x * 8 + wave;       // 0..1999
    const int n0    = ntile * 16;
    const int half  = lane >> 4;                   // K sub-pair select
    const int l15   = lane & 15;

    v8f acc[8];
#pragma unroll
    for (int m = 0; m < 8; ++m) acc[m] = v8f{};

    const float* WpRow = Wp + (size_t)(n0 + l15) * HIDDEN;
    const float* WlRow = Wl + (size_t)(n0 + l15) * HIDDEN;

    for (int kb = 0; kb < HIDDEN / KCH; ++kb) {
        // ---- cooperative async stage of the Cp/Cl K-chunk into LDS ----
#pragma unroll
        for (int rr = 0; rr < 4; ++rr) {
            const int li  = rr * 256 + tid;        // 0..1023 16B-quads per matrix
            const int row = li >> 3;               // 8 quads per 32-col row
            const int c4  = (li & 7) * 4;
            const unsigned loff0 =
                (unsigned)(uintptr_t)&sA[0][row * ASTRIDE + c4];
            const unsigned loff1 =
                (unsigned)(uintptr_t)&sA[1][row * ASTRIDE + c4];
            async_copy16_to_lds(loff0, Cp + (size_t)row * HIDDEN + kb * KCH + c4);
            async_copy16_to_lds(loff1, Cl + (size_t)row * HIDDEN + kb * KCH + c4);
        }
        wait_async0();
        __syncthreads();

        // ---- WMMA over this K-chunk: 2 passes x 8 slices x 8 M-tiles ----
#pragma unroll
        for (int pass = 0; pass < 2; ++pass) {
            const float* W  = (pass == 0) ? WpRow : WlRow;
            const float* sC = sA[pass];
#pragma unroll
            for (int ks = 0; ks < KCH; ks += 4) {
                v2f b = *(const v2f*)(W + kb * KCH + ks + half * 2);
#pragma unroll
                for (int m = 0; m < 8; ++m) {
                    v2f a = *(const v2f*)(sC + (m * 16 + l15) * ASTRIDE + ks + half * 2);
                    acc[m] = __builtin_amdgcn_wmma_f32_16x16x4_f32(
                        false, a, false, b, (short)0, acc[m], false, false);
                }
            }
        }
        __syncthreads();   // protect LDS before next chunk's async writes
    }

    // ---- writeback: per-row bias select + scatter to out[op[t]] ----
    const int n = n0 + l15;
#pragma unroll
    for (int m = 0; m < 8; ++m) {
#pragma unroll
        for (int r = 0; r < 8; ++r) {
            const int t = m * 16 + r + half * 8;
            const int k = kind[t];
            if (k != 0) {
                const float bias = (k == 2) ? bp[n] : bl[n];
                out[(size_t)opArr[t] * VOCAB + n] = acc[m][r] + bias;
            }
        }
    }
}

// ---------------------------------------------------------------------------
extern "C" void kernel_launch(void* const* d_in, const int* in_sizes, int n_in,
                              void* d_out, int out_size, void* d_ws, size_t ws_size,
                              hipStream_t stream) {
    const float* embed = (const float*)d_in[0];
    const float* Wpar  = (const float*)d_in[1];
    const float* bpar  = (const float*)d_in[2];
    const float* Wmu   = (const float*)d_in[3];
    const float* bmu   = (const float*)d_in[4];
    const float* Wlv   = (const float*)d_in[5];
    const float* blv   = (const float*)d_in[6];
    const float* W1    = (const float*)d_in[7];
    const float* b1    = (const float*)d_in[8];
    const float* W2    = (const float*)d_in[9];
    const float* b2    = (const float*)d_in[10];
    const float* Wl    = (const float*)d_in[11];
    const float* bl    = (const float*)d_in[12];
    const float* Wp    = (const float*)d_in[13];
    const float* bp    = (const float*)d_in[14];
    const float* eps   = (const float*)d_in[15];
    const int* values  = (const int*)d_in[16];
    const int* left    = (const int*)d_in[17];
    const int* right   = (const int*)d_in[18];

    float* out = (float*)d_out;       // [128*32000] out | [128] mu | [128] logvar
    float* ws  = (float*)d_ws;

    // Workspace layout (floats)
    float* h    = ws;                 // 511*256 = 130816
    float* z    = ws + 131072;        // 128
    float* Cp   = ws + 132096;        // 128*256
    float* Cl   = ws + 164864;        // 128*256
    int*   kind = (int*)(ws + 197632);// 128
    int*   opA  = kind + 128;         // 128

    const int OUT_TOTAL = MAX_STEPS * VOCAB + 2 * LATENT;   // 4096256

    // 0) zero output (inactive rows must be zero)
    k_zero<<<(OUT_TOTAL + 255) / 256, 256, 0, stream>>>(out, OUT_TOTAL);

    // 1) encoder: level-parallel bottom-up sweep (9 launches, serialized)
    for (int lvl = 8; lvl >= 0; --lvl) {
        int first = (1 << lvl) - 1;
        int last  = (1 << (lvl + 1)) - 1;
        if (last > N_NODES) last = N_NODES;
        int count = last - first;
        k_enc_level<<<count, 256, 0, stream>>>(embed, values, left, right,
                                               Wpar, bpar, h, first);
    }

    // 2) latent head (mu/logvar into d_out tail, z into ws)
    k_latent<<<1, 256, 0, stream>>>(h, Wmu, bmu, Wlv, blv, eps,
                                    out + MAX_STEPS * VOCAB,
                                    out + MAX_STEPS * VOCAB + LATENT, z);

    // 3) sequential control chain: records Cp/Cl/kind/op
    k_dec_control<<<1, 256, 0, stream>>>(W1, b1, W2, b2,
                                         Wp + (size_t)VOCAB * HIDDEN, bp + VOCAB,
                                         z, Cp, Cl, kind, opA);

    // 4) vocab GEMM via V_WMMA_F32_16X16X4_F32 + async-LDS A staging
    //    2000 n-tiles / 8 waves per block = 250 blocks
    k_vocab_gemm<<<250, 256, 0, stream>>>(Wl, bl, Wp, bp, Cp, Cl, kind, opA, out);
}